// Pendulum_True_Dynamics_10350871183738
// MI455X (gfx1250) — compile-verified
//
#include <hip/hip_runtime.h>
#include <hip/hip_bf16.h>
#include <math.h>

// Pendulum Dormand-Prince(5) fixed-grid scan.
// One trajectory per lane (wave32). B = 32768 -> 1024 waves.
// Transcendental-bound; atan2/fmod removed via angle-addition identity.

namespace {

constexpr float kMaxSpeed  = 8.0f;
constexpr float kMaxTorque = 2.0f;
// acc = 3*G/(2*L)*sin_th + 3/(M*L^2)*u with G=10, M=1, L=1:
constexpr float kSinGain = 15.0f;
constexpr float kUGain   = 3.0f;

__device__ __forceinline__ float clampf(float x, float lo, float hi) {
  return fminf(fmaxf(x, lo), hi);
}

// One dynamics evaluation f(t, y).
// searchsorted(times, t, 'right')-1 == i unless t >= times[i+1] (only reachable
// at the k6 stage t0+h); then idx==i+1 and (u,dt) switch to the lookahead pair.
// newth = atan2(y1, y0) + ntd*dt; we only need sin/cos(newth), so use:
//   sin(th+a) = (y1*cos a + y0*sin a)/r,  cos(th+a) = (y0*cos a - y1*sin a)/r,
//   r = sqrt(y0^2 + y1^2)   (angle_normalize is a no-op mod 2*pi)
// -> 1 v_rsq + 1 fast sincos per stage instead of atan2 + fmod + sincos.
__device__ __forceinline__ void dyn(float t, const float y[3],
                                    float t1, float u_cur, float u_nxt,
                                    float dt_cur, float dt_nxt,
                                    float k[3]) {
  const bool  nxt = (t >= t1);
  const float u   = nxt ? u_nxt : u_cur;
  const float dt  = nxt ? dt_nxt : dt_cur;

  const float acc = fmaf(kSinGain, y[1], kUGain * u);
  const float ntd = clampf(fmaf(acc, dt, y[2]), -kMaxSpeed, kMaxSpeed);

  const float a = ntd * dt;          // |a| <= 8*0.05 = 0.4 -> cheap sincos path
  float s, c;
  __sincosf(a, &s, &c);
  const float rinv = rsqrtf(fmaf(y[0], y[0], y[1] * y[1]));
  const float sn = fmaf(y[1], c,  y[0] * s) * rinv;   // sin(newth)
  const float cn = fmaf(y[0], c, -y[1] * s) * rinv;   // cos(newth)

  k[0] = -sn * ntd;
  k[1] =  cn * ntd;
  k[2] =  acc;
}

} // namespace

__global__ __launch_bounds__(256)
void pendulum_dopri5(const float* __restrict__ ob,
                     const float* __restrict__ acs,
                     const float* __restrict__ times,
                     float* __restrict__ out,
                     int B, int T) {
  const int b = blockIdx.x * blockDim.x + threadIdx.x;
  if (b >= B) return;

  const float* tm = times + (size_t)b * T;       // (B, T)
  const float* ac = acs   + (size_t)b * T;       // (B, T, 1)
  float*       o  = out   + (size_t)b * T * 3;   // (B, T, 3)

  float y[3] = { ob[3 * b + 0], ob[3 * b + 1], ob[3 * b + 2] };
  o[0] = y[0]; o[1] = y[1]; o[2] = y[2];

  // Carried scan state: t0=times[i], t1=times[i+1], clipped actions u0,u1.
  float t0 = tm[0];
  float t1 = tm[1];
  float u0 = clampf(ac[0], -kMaxTorque, kMaxTorque);
  float u1 = clampf(ac[1], -kMaxTorque, kMaxTorque);

  for (int i = 0; i < T - 1; ++i) {
    // Stream-prefetch ahead of the scan (lowers to global_prefetch_b8).
    int pf = i + 16; if (pf > T - 1) pf = T - 1;
    __builtin_prefetch(tm + pf, 0, 0);
    __builtin_prefetch(ac + pf, 0, 0);

    const float h    = t1 - t0;
    const bool  has2 = (i + 2 < T);
    const float t2   = has2 ? tm[i + 2] : t1;
    const float u2   = has2 ? clampf(ac[i + 2], -kMaxTorque, kMaxTorque) : u1;
    const float dtn  = t2 - t1;   // dt when idx lands on i+1 (0 at final step,
                                  // matching JAX's clamped OOB gather)

    float k1[3], k2[3], k3[3], k4[3], k5[3], k6[3], yt[3];

    dyn(t0, y, t1, u0, u1, h, dtn, k1);

#pragma unroll
    for (int j = 0; j < 3; ++j)
      yt[j] = fmaf(h, (1.0f/5.0f) * k1[j], y[j]);
    dyn(t0 + h * (1.0f/5.0f), yt, t1, u0, u1, h, dtn, k2);

#pragma unroll
    for (int j = 0; j < 3; ++j)
      yt[j] = fmaf(h, fmaf(3.0f/40.0f, k1[j], (9.0f/40.0f) * k2[j]), y[j]);
    dyn(t0 + h * (3.0f/10.0f), yt, t1, u0, u1, h, dtn, k3);

#pragma unroll
    for (int j = 0; j < 3; ++j)
      yt[j] = fmaf(h, fmaf(44.0f/45.0f, k1[j],
                   fmaf(-56.0f/15.0f, k2[j], (32.0f/9.0f) * k3[j])), y[j]);
    dyn(t0 + h * (4.0f/5.0f), yt, t1, u0, u1, h, dtn, k4);

#pragma unroll
    for (int j = 0; j < 3; ++j)
      yt[j] = fmaf(h, fmaf(19372.0f/6561.0f, k1[j],
                   fmaf(-25360.0f/2187.0f, k2[j],
                   fmaf(64448.0f/6561.0f, k3[j], (-212.0f/729.0f) * k4[j]))), y[j]);
    dyn(t0 + h * (8.0f/9.0f), yt, t1, u0, u1, h, dtn, k5);

#pragma unroll
    for (int j = 0; j < 3; ++j)
      yt[j] = fmaf(h, fmaf(9017.0f/3168.0f, k1[j],
                   fmaf(-355.0f/33.0f, k2[j],
                   fmaf(46732.0f/5247.0f, k3[j],
                   fmaf(49.0f/176.0f, k4[j], (-5103.0f/18656.0f) * k5[j])))), y[j]);
    dyn(t0 + h, yt, t1, u0, u1, h, dtn, k6);

#pragma unroll
    for (int j = 0; j < 3; ++j)
      y[j] = fmaf(h, fmaf(35.0f/384.0f, k1[j],
                  fmaf(500.0f/1113.0f, k3[j],
                  fmaf(125.0f/192.0f, k4[j],
                  fmaf(-2187.0f/6784.0f, k5[j], (11.0f/84.0f) * k6[j])))), y[j]);

    float* os = o + 3 * (i + 1);
    os[0] = y[0]; os[1] = y[1]; os[2] = y[2];

    // Slide the scan window.
    t0 = t1; t1 = t2;
    u0 = u1; u1 = u2;
  }
}

extern "C" void kernel_launch(void* const* d_in, const int* in_sizes, int n_in,
                              void* d_out, int out_size, void* d_ws, size_t ws_size,
                              hipStream_t stream) {
  const float* ob    = (const float*)d_in[0];   // (B, 3)
  const float* acs   = (const float*)d_in[1];   // (B, T, 1)
  const float* times = (const float*)d_in[2];   // (B, T)
  float* out = (float*)d_out;                   // (B, T, 3)

  const int B = in_sizes[0] / 3;
  const int T = in_sizes[2] / B;

  const int block = 256;                        // 8 wave32s per workgroup
  const int grid  = (B + block - 1) / block;
  pendulum_dopri5<<<grid, block, 0, stream>>>(ob, acs, times, out, B, T);
}